// GAT_6373731467288
// MI455X (gfx1250) — compile-verified
//
#include <hip/hip_runtime.h>
#include <math.h>

typedef __attribute__((ext_vector_type(2))) float v2f;
typedef __attribute__((ext_vector_type(8))) float v8f;

#define HEADS 8
#define HID 32
#define HD 256          // HEADS*HID
#define NEG_SLOPE 0.2f
#define NSUB 4          // max N-tiles (16 cols each) per wave -> 64-wide strip

// ---------------------------------------------------------------------------
// fp32 WMMA GEMM: C[M x NN] = A[M x KK] @ B[KK x NN] (+ bias).
// Fully compile-time shape: every load is base + immediate offset
// (global_load, LOADcnt only, no induction arithmetic), K loop fully
// unrolled so loads cluster ahead of the V_WMMA_F32_16X16X4_F32 chain.
// One wave per 16 x (16*NACT) strip; A fragment shared by NACT accumulators.
// M must be a multiple of 16. RAGGED adds branchless column masking
// (EXEC stays all-ones at every WMMA either way).
// ---------------------------------------------------------------------------
template <int NN, int KK, bool RAGGED>
__global__ void gat_wmma_gemm_f32(const float* __restrict__ A,
                                  const float* __restrict__ B,
                                  const float* __restrict__ bias,
                                  float* __restrict__ C,
                                  int tiles_m, int ngroups) {
    constexpr int TILES_N = (NN + 15) / 16;
    constexpr int NACT = TILES_N < NSUB ? TILES_N : NSUB;  // live subtiles

    int wave = blockIdx.x * (blockDim.x >> 5) + (threadIdx.x >> 5);
    if (wave >= tiles_m * ngroups) return;
    int tm = wave % tiles_m;
    int tg = wave / tiles_m;
    int m0 = tm << 4;
    int n0 = tg << 6;               // 64 columns per group

    int lane = threadIdx.x & 31;
    int half = lane >> 4;           // 0: lanes 0-15, 1: lanes 16-31
    int l15  = lane & 15;
    int koff = half << 1;           // fragment K sub-offset {0,2}
    int arow = m0 + l15;            // always in range (M % 16 == 0)

    // Per-lane 32-bit element offsets into B; all k/row deltas are
    // compile-time constants folded into load immediates.
    float bmask[NACT];
    unsigned boff[NACT];
#pragma unroll
    for (int j = 0; j < NACT; ++j) {
        int bcol = n0 + (j << 4) + l15;
        int bcolc = bcol;
        if (RAGGED) {
            bmask[j] = (bcol < NN) ? 1.0f : 0.0f;
            bcolc = (bcol < NN) ? bcol : (NN - 1);
        }
        boff[j] = (unsigned)(koff * NN + bcolc);
    }
    const float* ap = A + (size_t)arow * KK + koff;

    v8f acc[NACT];
#pragma unroll
    for (int j = 0; j < NACT; ++j) acc[j] = (v8f){};

#pragma unroll
    for (int k = 0; k < KK; k += 4) {
        // ---- load phase: 1x b64 (A) + 2*NACT x b32 (B), immediate offsets
        v2f a; a.x = ap[k]; a.y = ap[k + 1];
        float t0[NACT], t1[NACT];
#pragma unroll
        for (int j = 0; j < NACT; ++j) {
            t0[j] = B[boff[j] + k * NN];
            t1[j] = B[boff[j] + k * NN + NN];
        }
        // ---- compute phase: NACT WMMAs sharing the A fragment ----
#pragma unroll
        for (int j = 0; j < NACT; ++j) {
            v2f b;
            if (RAGGED) { b.x = t0[j] * bmask[j]; b.y = t1[j] * bmask[j]; }
            else        { b.x = t0[j];            b.y = t1[j]; }
            acc[j] = __builtin_amdgcn_wmma_f32_16x16x4_f32(
                false, a, false, b, (short)0, acc[j], false, false);
        }
    }

#pragma unroll
    for (int j = 0; j < NACT; ++j) {
        int bcol = n0 + (j << 4) + l15;
        if (!RAGGED || bcol < NN) {
            float bv = bias ? bias[bcol] : 0.0f;
#pragma unroll
            for (int i = 0; i < 8; ++i) {
                int row = m0 + i + (half << 3);  // VGPR i: M=i (lo), M=i+8 (hi)
                C[(size_t)row * NN + bcol] = acc[j][i] + bv;
            }
        }
    }
}

// ---------------------------------------------------------------------------
// alpha_s / alpha_d: per (node, head) dot of xh[n,h,:] with a_src/a_dst[h,:]
// ---------------------------------------------------------------------------
__global__ void gat_alpha_k(const float* __restrict__ xh,
                            const float* __restrict__ a_src,
                            const float* __restrict__ a_dst,
                            float* __restrict__ as, float* __restrict__ ad,
                            int N) {
    int t = blockIdx.x * blockDim.x + threadIdx.x;
    if (t >= N * HEADS) return;
    int n = t >> 3, h = t & 7;
    const float* p  = xh + (size_t)n * HD + h * HID;
    const float* ps = a_src + h * HID;
    const float* pd = a_dst + h * HID;
    float ss = 0.f, sd = 0.f;
#pragma unroll
    for (int i = 0; i < HID; ++i) {
        float v = p[i];
        ss += v * ps[i];
        sd += v * pd[i];
    }
    as[t] = ss;
    ad[t] = sd;
}

__global__ void gat_init_softmax(float* __restrict__ emax,
                                 float* __restrict__ denom, int n) {
    int i = blockIdx.x * blockDim.x + threadIdx.x;
    if (i < n) { emax[i] = -1e30f; denom[i] = 0.0f; }
}

__device__ __forceinline__ void atomicMaxFloat(float* addr, float val) {
    // IEEE bit-trick max (works for finite values; init is -1e30)
    if (val >= 0.0f)
        atomicMax((int*)addr, __float_as_int(val));
    else
        atomicMin((unsigned int*)addr, (unsigned int)__float_as_int(val));
}

__device__ __forceinline__ float leaky(float e) {
    return e > 0.0f ? e : NEG_SLOPE * e;
}

// Edge index extension: e in [0,E) -> edge list; e in [E, E+N) -> self loop.
__device__ __forceinline__ void edge_sd(const int* __restrict__ src,
                                        const int* __restrict__ dst,
                                        int e, int E, int& s, int& d) {
    if (e < E) { s = src[e]; d = dst[e]; }
    else       { s = e - E; d = s; }
}

__global__ void gat_edge_max(const int* __restrict__ src, const int* __restrict__ dst,
                             int E, int Etot,
                             const float* __restrict__ as, const float* __restrict__ ad,
                             float* __restrict__ emax) {
    int t = blockIdx.x * blockDim.x + threadIdx.x;
    if (t >= Etot * HEADS) return;
    int e = t >> 3, h = t & 7;
    int s, d; edge_sd(src, dst, e, E, s, d);
    float ev = leaky(as[s * HEADS + h] + ad[d * HEADS + h]);
    atomicMaxFloat(&emax[d * HEADS + h], ev);
}

__global__ void gat_edge_expsum(const int* __restrict__ src, const int* __restrict__ dst,
                                int E, int Etot,
                                const float* __restrict__ as, const float* __restrict__ ad,
                                const float* __restrict__ emax,
                                float* __restrict__ denom) {
    int t = blockIdx.x * blockDim.x + threadIdx.x;
    if (t >= Etot * HEADS) return;
    int e = t >> 3, h = t & 7;
    int s, d; edge_sd(src, dst, e, E, s, d);
    float ev = leaky(as[s * HEADS + h] + ad[d * HEADS + h]);
    float ex = __expf(ev - emax[d * HEADS + h]);
    atomicAdd(&denom[d * HEADS + h], ex);
}

// One wave per edge. Lanes 0-7 compute the 8 head weights; broadcast via
// shfl; 8 coalesced 128B gathers from xh[src] and atomic adds into out[dst].
__global__ void gat_edge_agg(const int* __restrict__ src, const int* __restrict__ dst,
                             int E, int Etot,
                             const float* __restrict__ as, const float* __restrict__ ad,
                             const float* __restrict__ emax,
                             const float* __restrict__ denom,
                             const float* __restrict__ xh,
                             float* __restrict__ out) {
    int wave = blockIdx.x * (blockDim.x >> 5) + (threadIdx.x >> 5);
    if (wave >= Etot) return;
    int lane = threadIdx.x & 31;
    int s, d; edge_sd(src, dst, wave, E, s, d);

    float w = 0.0f;
    if (lane < HEADS) {
        float ev = leaky(as[s * HEADS + lane] + ad[d * HEADS + lane]);
        w = __expf(ev - emax[d * HEADS + lane]) / denom[d * HEADS + lane];
    }
    const float* xs = xh + (size_t)s * HD;
    float* od = out + (size_t)d * HD;
#pragma unroll
    for (int h = 0; h < HEADS; ++h) {
        float wh = __shfl(w, h, 32);
        atomicAdd(&od[h * HID + lane], wh * xs[h * HID + lane]);
    }
}

__global__ void gat_bias_relu(float* __restrict__ out, const float* __restrict__ b,
                              int total) {
    int i = blockIdx.x * blockDim.x + threadIdx.x;
    if (i < total) {
        float v = out[i] + b[i & (HD - 1)];
        out[i] = v > 0.0f ? v : 0.0f;
    }
}

// ---------------------------------------------------------------------------
extern "C" void kernel_launch(void* const* d_in, const int* in_sizes, int n_in,
                              void* d_out, int out_size, void* d_ws, size_t ws_size,
                              hipStream_t stream) {
    const float* x   = (const float*)d_in[0];
    const int*   ei  = (const int*)d_in[1];
    const float* W1  = (const float*)d_in[2];
    const float* a1s = (const float*)d_in[3];
    const float* a1d = (const float*)d_in[4];
    const float* b1  = (const float*)d_in[5];
    const float* W2  = (const float*)d_in[6];
    const float* a2s = (const float*)d_in[7];
    const float* a2d = (const float*)d_in[8];
    const float* b2  = (const float*)d_in[9];
    const float* Wc  = (const float*)d_in[10];
    const float* bc  = (const float*)d_in[11];

    const int N = in_sizes[0] / 128;   // 50000
    const int E = in_sizes[1] / 2;     // 800000
    const int Etot = E + N;            // + self loops
    const int* src = ei;
    const int* dst = ei + E;

    float* ws    = (float*)d_ws;
    float* xh    = ws;                          // N*256
    float* hbuf  = xh   + (size_t)N * HD;       // N*256
    float* as_   = hbuf + (size_t)N * HD;       // N*8
    float* ad_   = as_  + (size_t)N * HEADS;    // N*8
    float* emax  = ad_  + (size_t)N * HEADS;    // N*8
    float* denom = emax + (size_t)N * HEADS;    // N*8

    const int BT = 256;
    const int tiles_m = N / 16;
    auto cdiv = [](int a, int b) { return (a + b - 1) / b; };
    const int ngroups = HD / (16 * NSUB);       // 4 groups of 64 cols
    const int gemm_blocks = cdiv(tiles_m * ngroups, 8);

    // ---------------- Layer 1 ----------------
    gat_wmma_gemm_f32<HD, 128, false><<<gemm_blocks, BT, 0, stream>>>(
        x, W1, nullptr, xh, tiles_m, ngroups);
    gat_alpha_k<<<cdiv(N * HEADS, BT), BT, 0, stream>>>(xh, a1s, a1d, as_, ad_, N);
    gat_init_softmax<<<cdiv(N * HEADS, BT), BT, 0, stream>>>(emax, denom, N * HEADS);
    hipMemsetAsync(hbuf, 0, (size_t)N * HD * sizeof(float), stream);
    gat_edge_max<<<cdiv(Etot * HEADS, BT), BT, 0, stream>>>(src, dst, E, Etot, as_, ad_, emax);
    gat_edge_expsum<<<cdiv(Etot * HEADS, BT), BT, 0, stream>>>(src, dst, E, Etot, as_, ad_, emax, denom);
    gat_edge_agg<<<cdiv(Etot, 8), BT, 0, stream>>>(src, dst, E, Etot, as_, ad_, emax, denom, xh, hbuf);
    gat_bias_relu<<<cdiv(N * HD, BT), BT, 0, stream>>>(hbuf, b1, N * HD);

    // ---------------- Layer 2 ----------------
    gat_wmma_gemm_f32<HD, HD, false><<<gemm_blocks, BT, 0, stream>>>(
        hbuf, W2, nullptr, xh, tiles_m, ngroups);
    gat_alpha_k<<<cdiv(N * HEADS, BT), BT, 0, stream>>>(xh, a2s, a2d, as_, ad_, N);
    gat_init_softmax<<<cdiv(N * HEADS, BT), BT, 0, stream>>>(emax, denom, N * HEADS);
    // hbuf reused as accumulator; GEMM above already consumed it (stream order)
    hipMemsetAsync(hbuf, 0, (size_t)N * HD * sizeof(float), stream);
    gat_edge_max<<<cdiv(Etot * HEADS, BT), BT, 0, stream>>>(src, dst, E, Etot, as_, ad_, emax);
    gat_edge_expsum<<<cdiv(Etot * HEADS, BT), BT, 0, stream>>>(src, dst, E, Etot, as_, ad_, emax, denom);
    gat_edge_agg<<<cdiv(Etot, 8), BT, 0, stream>>>(src, dst, E, Etot, as_, ad_, emax, denom, xh, hbuf);
    gat_bias_relu<<<cdiv(N * HD, BT), BT, 0, stream>>>(hbuf, b2, N * HD);

    // ---------------- Classifier: d_out = h2 @ Wc + bc (N x 256 @ 256 x 40)
    gat_wmma_gemm_f32<40, HD, true><<<cdiv(tiles_m, 8), BT, 0, stream>>>(
        hbuf, Wc, bc, (float*)d_out, tiles_m, /*ngroups=*/1);
}